// MonolithicWrapper_58093727646225
// MI455X (gfx1250) — compile-verified
//
#include <hip/hip_runtime.h>
#include <hip/hip_bf16.h>

#define L_ 8
#define H_ 1024
#define NH_ 16
#define KVH_ 4
#define HD_ 64
#define INTER_ 2816
#define VOCAB_ 32000
#define CTX_ 2048

typedef __attribute__((ext_vector_type(2))) float v2f;
typedef __attribute__((ext_vector_type(8))) float v8f;
typedef __attribute__((ext_vector_type(4))) int   v4i;

// Pointer-to-v4i in explicit address spaces for the async LDS copy builtin:
// param0 = global (addrspace(1), printed as "__device__"), param1 = LDS (addrspace(3)).
typedef __attribute__((address_space(1))) v4i* gv4i_p;
typedef __attribute__((address_space(3))) v4i* lv4i_p;

// ---------------------------------------------------------------------------
// Embedding gather: h[i] = embed[ids[0]*H + i]   (1 block, 256 threads)
// ---------------------------------------------------------------------------
__global__ __launch_bounds__(256) void embed_kernel(const float* __restrict__ embed,
                                                    const int* __restrict__ ids,
                                                    float* __restrict__ h) {
    int tid = threadIdx.x;
    const float* src = embed + (size_t)ids[0] * H_;
#pragma unroll
    for (int j = 0; j < H_ / 256; j++) h[tid + j * 256] = src[tid + j * 256];
}

// ---------------------------------------------------------------------------
// Fused: h_out = base + sum_s part[s]  (optional), then RMSNorm -> x_out.
// Single block, 256 threads, n = H_ = 1024 (4 elements/thread in registers).
// ---------------------------------------------------------------------------
__global__ __launch_bounds__(256) void fused_norm_kernel(const float* __restrict__ base,
                                                         const float* __restrict__ part, int S,
                                                         const float* __restrict__ w,
                                                         float* __restrict__ h_out,
                                                         float* __restrict__ x_out) {
    __shared__ float red[256];
    int tid = threadIdx.x;
    float hv[4];
    float ss = 0.f;
#pragma unroll
    for (int j = 0; j < 4; j++) {
        int i = tid + j * 256;
        float b = base[i];
        if (part) {
            for (int s = 0; s < S; s++) b += part[(size_t)s * H_ + i];
        }
        hv[j] = b;
        if (h_out) h_out[i] = b;
        ss += b * b;
    }
    red[tid] = ss;
    __syncthreads();
    for (int off = 128; off > 0; off >>= 1) {
        if (tid < off) red[tid] += red[tid + off];
        __syncthreads();
    }
    float inv = rsqrtf(red[0] / (float)H_ + 1e-5f);
#pragma unroll
    for (int j = 0; j < 4; j++) {
        int i = tid + j * 256;
        x_out[i] = hv[j] * inv * w[i];
    }
}

// ---------------------------------------------------------------------------
// Core: 16 output rows per wave via V_WMMA_F32_16X16X4_F32, x broadcast in B.
// A 16x4 f32 layout: lane<16 -> K=0,1 ; lane>=16 -> K=2,3 (mirrored for B).
// D col 0: lane0 -> M=0..7 in c[0..7], lane16 -> M=8..15.
// Weights stream from HBM (b64/lane + prefetch), x comes from LDS.
// ---------------------------------------------------------------------------
__device__ __forceinline__ void gemv16_wmma(const float* __restrict__ Wbase,  // W + rowBase*K
                                            int K, int k0, int klen,
                                            const float* xs,                  // LDS, local idx
                                            float* __restrict__ po, int lane) {
    int m = lane & 15;
    int ksel = (lane >> 4) << 1;                 // 0 or 2
    const float* wp = Wbase + (size_t)m * K + k0 + ksel;
    const float* xp = xs + ksel;
    v8f c = {};
    for (int kk = 0; kk < klen; kk += 16) {
        __builtin_prefetch(wp + kk + 128, 0, 0);       // global_prefetch stream hint
        v2f a0 = *(const v2f*)(wp + kk);
        v2f b0 = *(const v2f*)(xp + kk);
        v2f a1 = *(const v2f*)(wp + kk + 4);
        v2f b1 = *(const v2f*)(xp + kk + 4);
        v2f a2 = *(const v2f*)(wp + kk + 8);
        v2f b2 = *(const v2f*)(xp + kk + 8);
        v2f a3 = *(const v2f*)(wp + kk + 12);
        v2f b3 = *(const v2f*)(xp + kk + 12);
        c = __builtin_amdgcn_wmma_f32_16x16x4_f32(false, a0, false, b0, (short)0, c, false, false);
        c = __builtin_amdgcn_wmma_f32_16x16x4_f32(false, a1, false, b1, (short)0, c, false, false);
        c = __builtin_amdgcn_wmma_f32_16x16x4_f32(false, a2, false, b2, (short)0, c, false, false);
        c = __builtin_amdgcn_wmma_f32_16x16x4_f32(false, a3, false, b3, (short)0, c, false, false);
    }
    if (lane == 0) {
#pragma unroll
        for (int i = 0; i < 8; i++) po[i] = c[i];
    } else if (lane == 16) {
#pragma unroll
        for (int i = 0; i < 8; i++) po[8 + i] = c[i];
    }
}

// Stage x[k0 .. k0+klen) into LDS with async copies (ASYNCcnt path), klen%16==0.
__device__ __forceinline__ void stage_x_async(const float* __restrict__ x, int k0, int klen,
                                              float* xs /* shared, >= klen floats */) {
    int t = threadIdx.x;
    int nvec = klen >> 2;                        // 16B transfers
    if (t < nvec) {
        float* gp = const_cast<float*>(x) + k0 + t * 4;
        __builtin_amdgcn_global_load_async_to_lds_b128((gv4i_p)gp, (lv4i_p)(xs + t * 4), 0, 0);
    }
    asm volatile("s_wait_asynccnt 0x0" ::: "memory");
    __syncthreads();
}

// ---------------------------------------------------------------------------
// GEMV kernels (split-K partials): grid = (M/128, S); 256 threads (8 waves).
// part[seg*M + r] = sum_{k in seg} W[r,k] x[k].
// ---------------------------------------------------------------------------
__global__ __launch_bounds__(256) void gemv_single_kernel(const float* __restrict__ W,
                                                          const float* __restrict__ x,
                                                          float* __restrict__ part,
                                                          int M, int K) {
    __shared__ float xs[1024];
    int S = gridDim.y;
    int seg = blockIdx.y;
    int klen = K / S;
    int k0 = seg * klen;
    stage_x_async(x, k0, klen, xs);
    int rowBase = (blockIdx.x * 8 + (threadIdx.x >> 5)) * 16;
    gemv16_wmma(W + (size_t)rowBase * K, K, k0, klen, xs,
                part + (size_t)seg * M + rowBase, threadIdx.x & 31);
}

// Fused q|k|v projection: rows [0,1024) q_w, [1024,1280) k_w, [1280,1536) v_w.
__global__ __launch_bounds__(256) void gemv_qkv_kernel(const float* __restrict__ qw,
                                                       const float* __restrict__ kw,
                                                       const float* __restrict__ vw,
                                                       const float* __restrict__ x,
                                                       float* __restrict__ part) {
    __shared__ float xs[1024];
    const int K = H_, M = NH_ * HD_ + 2 * KVH_ * HD_;   // 1536
    int S = gridDim.y;
    int seg = blockIdx.y;
    int klen = K / S;
    int k0 = seg * klen;
    stage_x_async(x, k0, klen, xs);
    int rowBase = (blockIdx.x * 8 + (threadIdx.x >> 5)) * 16;   // uniform per wave
    const float* Wb;
    if (rowBase < NH_ * HD_)                   Wb = qw + (size_t)rowBase * K;
    else if (rowBase < NH_ * HD_ + KVH_ * HD_) Wb = kw + (size_t)(rowBase - NH_ * HD_) * K;
    else                                       Wb = vw + (size_t)(rowBase - NH_ * HD_ - KVH_ * HD_) * K;
    gemv16_wmma(Wb, K, k0, klen, xs, part + (size_t)seg * M + rowBase, threadIdx.x & 31);
}

// Fused gate|up projection: rows [0,2816) gate_w, [2816,5632) up_w.
__global__ __launch_bounds__(256) void gemv_gateup_kernel(const float* __restrict__ gw,
                                                          const float* __restrict__ uw,
                                                          const float* __restrict__ x,
                                                          float* __restrict__ part) {
    __shared__ float xs[1024];
    const int K = H_, M = 2 * INTER_;                   // 5632
    int S = gridDim.y;
    int seg = blockIdx.y;
    int klen = K / S;
    int k0 = seg * klen;
    stage_x_async(x, k0, klen, xs);
    int rowBase = (blockIdx.x * 8 + (threadIdx.x >> 5)) * 16;
    const float* Wb = (rowBase < INTER_) ? gw + (size_t)rowBase * K
                                         : uw + (size_t)(rowBase - INTER_) * K;
    gemv16_wmma(Wb, K, k0, klen, xs, part + (size_t)seg * M + rowBase, threadIdx.x & 31);
}

// ---------------------------------------------------------------------------
// Fused split-K reduce + RoPE for q,k ; passthrough for v. 6 blocks x 256.
// part stride M=1536. RoPE partner (+/-32 within a 64-elem head) is in-block.
// ---------------------------------------------------------------------------
__global__ __launch_bounds__(256) void qkv_redrope_kernel(const float* __restrict__ part, int S,
                                                          const float* __restrict__ cosc,
                                                          const float* __restrict__ sinc,
                                                          const int* __restrict__ pos_ids,
                                                          float* __restrict__ qr,
                                                          float* __restrict__ kr,
                                                          float* __restrict__ vb) {
    __shared__ float sval[256];
    const int M = NH_ * HD_ + 2 * KVH_ * HD_;           // 1536
    int tid = threadIdx.x;
    int i = blockIdx.x * 256 + tid;
    float v = 0.f;
    for (int s = 0; s < S; s++) v += part[(size_t)s * M + i];
    sval[tid] = v;
    __syncthreads();
    int p = pos_ids[0];
    int d = i & (HD_ - 1);
    float cs = cosc[(size_t)p * HD_ + d];
    float sn = sinc[(size_t)p * HD_ + d];
    float rot = (d < HD_ / 2) ? -sval[tid + HD_ / 2] : sval[tid - HD_ / 2];
    if (i < NH_ * HD_) {
        qr[i] = v * cs + rot * sn;
    } else if (i < NH_ * HD_ + KVH_ * HD_) {
        kr[i - NH_ * HD_] = v * cs + rot * sn;
    } else {
        vb[i - NH_ * HD_ - KVH_ * HD_] = v;
    }
}

// ---------------------------------------------------------------------------
// GQA attention, one block per KV head (4 q heads), 256 threads.
// Scores in LDS (32KB); mask-based K/V row substitution (cache never written).
// ---------------------------------------------------------------------------
__global__ __launch_bounds__(256) void attn_kernel(const float* __restrict__ Kc,  // [KVH][CTX][HD]
                                                   const float* __restrict__ Vc,
                                                   const float* __restrict__ q,   // [16*64] roped
                                                   const float* __restrict__ knew,// [4*64] roped
                                                   const float* __restrict__ vnew,// [4*64]
                                                   const float* __restrict__ cmask,
                                                   const float* __restrict__ umask,
                                                   float* __restrict__ out) {     // [16*64]
    __shared__ float s[4][CTX_];
    __shared__ float qs[4][HD_];
    __shared__ float kn[HD_];
    __shared__ float red[4][64];
    __shared__ float hmax[4], hsum[4];

    int g   = blockIdx.x;
    int tid = threadIdx.x;

    if (tid < 4 * HD_) qs[tid >> 6][tid & 63] = q[(size_t)g * 4 * HD_ + tid];
    if (tid < HD_)     kn[tid] = knew[(size_t)g * HD_ + tid];
    __syncthreads();

    const float scale = 0.125f;   // 1/sqrt(64)

    for (int t = tid; t < CTX_; t += 256) {
        const float* Kt = Kc + ((size_t)g * CTX_ + t) * HD_;
        float um = umask[t];
        float cm = cmask[t];
        float d0 = 0.f, d1 = 0.f, d2 = 0.f, d3 = 0.f;
#pragma unroll 4
        for (int d = 0; d < HD_; d += 4) {
            float4 kv4 = *(const float4*)(Kt + d);
            float4 kn4 = *(const float4*)(&kn[d]);
            float b0 = kv4.x + um * (kn4.x - kv4.x);
            float b1 = kv4.y + um * (kn4.y - kv4.y);
            float b2 = kv4.z + um * (kn4.z - kv4.z);
            float b3 = kv4.w + um * (kn4.w - kv4.w);
            d0 += qs[0][d] * b0 + qs[0][d + 1] * b1 + qs[0][d + 2] * b2 + qs[0][d + 3] * b3;
            d1 += qs[1][d] * b0 + qs[1][d + 1] * b1 + qs[1][d + 2] * b2 + qs[1][d + 3] * b3;
            d2 += qs[2][d] * b0 + qs[2][d + 1] * b1 + qs[2][d + 2] * b2 + qs[2][d + 3] * b3;
            d3 += qs[3][d] * b0 + qs[3][d + 1] * b1 + qs[3][d + 2] * b2 + qs[3][d + 3] * b3;
        }
        s[0][t] = d0 * scale + cm;
        s[1][t] = d1 * scale + cm;
        s[2][t] = d2 * scale + cm;
        s[3][t] = d3 * scale + cm;
    }
    __syncthreads();

    int h  = tid >> 6;
    int lt = tid & 63;
    float mx = -3.4e38f;
    for (int t = lt; t < CTX_; t += 64) mx = fmaxf(mx, s[h][t]);
    red[h][lt] = mx;
    __syncthreads();
    if (lt == 0) {
        float m = red[h][0];
        for (int i = 1; i < 64; i++) m = fmaxf(m, red[h][i]);
        hmax[h] = m;
    }
    __syncthreads();
    float m = hmax[h];
    float sm = 0.f;
    for (int t = lt; t < CTX_; t += 64) {
        float p = __expf(s[h][t] - m);
        s[h][t] = p;
        sm += p;
    }
    red[h][lt] = sm;
    __syncthreads();
    if (lt == 0) {
        float ssum = 0.f;
        for (int i = 0; i < 64; i++) ssum += red[h][i];
        hsum[h] = ssum;
    }
    __syncthreads();

    float inv = 1.0f / hsum[h];
    float vn  = vnew[(size_t)g * HD_ + lt];
    float acc = 0.f;
    for (int t = 0; t < CTX_; t++) {
        float vv = Vc[((size_t)g * CTX_ + t) * HD_ + lt];
        float um = umask[t];
        acc += s[h][t] * (vv + um * (vn - vv));
    }
    out[((size_t)g * 4 + h) * HD_ + lt] = acc * inv;
}

// ---------------------------------------------------------------------------
// Fused split-K reduce + SiLU*mul: ib[i] = silu(sum g_part) * (sum u_part)
// part stride M = 2*INTER_ (gate rows first, then up rows).
// ---------------------------------------------------------------------------
__global__ __launch_bounds__(256) void gu_redsilu_kernel(const float* __restrict__ part, int S,
                                                         float* __restrict__ ib) {
    const int M = 2 * INTER_;
    int i = blockIdx.x * 256 + threadIdx.x;
    if (i >= INTER_) return;
    float g = 0.f, u = 0.f;
    for (int s = 0; s < S; s++) {
        g += part[(size_t)s * M + i];
        u += part[(size_t)s * M + INTER_ + i];
    }
    ib[i] = (g / (1.f + __expf(-g))) * u;
}

// ---------------------------------------------------------------------------
// Argmax (first-index tie-break, matching jnp.argmax), two stages.
// ---------------------------------------------------------------------------
__global__ __launch_bounds__(256) void argmax_part_kernel(const float* __restrict__ v, int n,
                                                          float* __restrict__ pm,
                                                          int* __restrict__ pi) {
    __shared__ float sm[256];
    __shared__ int   si[256];
    int tid = threadIdx.x;
    int i = blockIdx.x * 256 + tid;
    sm[tid] = (i < n) ? v[i] : -3.4e38f;
    si[tid] = i;
    __syncthreads();
    for (int off = 128; off > 0; off >>= 1) {
        if (tid < off) {
            float a = sm[tid], b = sm[tid + off];
            int ia = si[tid], ib = si[tid + off];
            if (b > a || (b == a && ib < ia)) { sm[tid] = b; si[tid] = ib; }
        }
        __syncthreads();
    }
    if (tid == 0) { pm[blockIdx.x] = sm[0]; pi[blockIdx.x] = si[0]; }
}

__global__ void argmax_final_kernel(const float* __restrict__ pm,
                                    const int* __restrict__ pi, int nb,
                                    float* __restrict__ out) {
    if (threadIdx.x == 0 && blockIdx.x == 0) {
        float best = pm[0];
        int bi = pi[0];
        for (int b = 1; b < nb; b++) {
            if (pm[b] > best || (pm[b] == best && pi[b] < bi)) { best = pm[b]; bi = pi[b]; }
        }
        out[0] = (float)bi;   // token_id
        out[1] = best;        // token_logit
    }
}

// ---------------------------------------------------------------------------
// Host-side orchestration
// ---------------------------------------------------------------------------
extern "C" void kernel_launch(void* const* d_in, const int* in_sizes, int n_in,
                              void* d_out, int out_size, void* d_ws, size_t ws_size,
                              hipStream_t stream) {
    const float* embed    = (const float*)d_in[0];
    const float* q_w      = (const float*)d_in[1];
    const float* k_w      = (const float*)d_in[2];
    const float* v_w      = (const float*)d_in[3];
    const float* o_w      = (const float*)d_in[4];
    const float* gate_w   = (const float*)d_in[5];
    const float* up_w     = (const float*)d_in[6];
    const float* down_w   = (const float*)d_in[7];
    const float* ln1_w    = (const float*)d_in[8];
    const float* ln2_w    = (const float*)d_in[9];
    const float* norm_w   = (const float*)d_in[10];
    const float* lm_head  = (const float*)d_in[11];
    const float* kv_cache = (const float*)d_in[12];
    const float* cos_c    = (const float*)d_in[13];
    const float* sin_c    = (const float*)d_in[14];
    const float* cmask    = (const float*)d_in[15];
    const float* umask    = (const float*)d_in[16];
    const int*   ids      = (const int*)d_in[17];
    const int*   pos_ids  = (const int*)d_in[18];

    float* ws = (float*)d_ws;
    float* h      = ws + 0;        // 1024
    float* h2     = ws + 1024;     // 1024
    float* x      = ws + 2048;     // 1024
    float* qr     = ws + 3072;     // 1024
    float* kr     = ws + 4096;     // 256
    float* vb     = ws + 4352;     // 256
    float* attnb  = ws + 4608;     // 1024
    float* ib     = ws + 5632;     // 2816
    float* logits = ws + 8448;     // 32000
    float* part   = ws + 40448;    // up to 8*5632 = 45056
    float* pm     = ws + 85504;    // 128
    int*   pi     = (int*)(ws + 85632);

    const int S = 8;                              // split-K segments
    const size_t KVL = (size_t)KVH_ * CTX_ * HD_;

    embed_kernel<<<1, 256, 0, stream>>>(embed, ids, h);
    fused_norm_kernel<<<1, 256, 0, stream>>>(h, nullptr, 0, ln1_w, nullptr, x);

    for (int l = 0; l < L_; l++) {
        // --- attention block ---
        gemv_qkv_kernel<<<dim3(1536 / 128, S), 256, 0, stream>>>(
            q_w + (size_t)l * H_ * H_,
            k_w + (size_t)l * KVH_ * HD_ * H_,
            v_w + (size_t)l * KVH_ * HD_ * H_, x, part);
        qkv_redrope_kernel<<<6, 256, 0, stream>>>(part, S, cos_c, sin_c, pos_ids, qr, kr, vb);

        attn_kernel<<<KVH_, 256, 0, stream>>>(
            kv_cache + (size_t)l * KVL, kv_cache + (size_t)(L_ + l) * KVL,
            qr, kr, vb, cmask, umask, attnb);

        gemv_single_kernel<<<dim3(H_ / 128, S), 256, 0, stream>>>(
            o_w + (size_t)l * H_ * H_, attnb, part, H_, H_);
        // h2 = h + o_proj ; x = rmsnorm(h2, ln2)
        fused_norm_kernel<<<1, 256, 0, stream>>>(h, part, S, ln2_w + (size_t)l * H_, h2, x);

        // --- MLP block ---
        gemv_gateup_kernel<<<dim3(2 * INTER_ / 128, S), 256, 0, stream>>>(
            gate_w + (size_t)l * INTER_ * H_, up_w + (size_t)l * INTER_ * H_, x, part);
        gu_redsilu_kernel<<<(INTER_ + 255) / 256, 256, 0, stream>>>(part, S, ib);

        gemv_single_kernel<<<dim3(H_ / 128, S), 256, 0, stream>>>(
            down_w + (size_t)l * H_ * INTER_, ib, part, H_, INTER_);
        // h = h2 + down ; x = rmsnorm(h, next ln1 or final norm)
        const float* wn = (l == L_ - 1) ? norm_w : ln1_w + (size_t)(l + 1) * H_;
        fused_norm_kernel<<<1, 256, 0, stream>>>(h2, part, S, wn, h, x);
    }

    // --- lm_head (S=1 writes logits directly) + argmax ---
    gemv_single_kernel<<<dim3(VOCAB_ / 128, 1), 256, 0, stream>>>(lm_head, x, logits, VOCAB_, H_);
    argmax_part_kernel<<<VOCAB_ / 256, 256, 0, stream>>>(logits, VOCAB_, pm, pi);
    argmax_final_kernel<<<1, 64, 0, stream>>>(pm, pi, VOCAB_ / 256, (float*)d_out);
}